// MaxDrawdownLoss_78950088835195
// MI455X (gfx1250) — compile-verified
//
#include <hip/hip_runtime.h>
#include <cstdint>

// MaxDrawdownLoss for MI455X (gfx1250, wave32).
// Memory-bound streaming scan: 67 MB in / 4 B out -> ~2.9 us HBM floor.
// Data path: async global->LDS DMA (ASYNCcnt), double buffered per wave.
// Compute path: wave32 shuffle prefix-product / prefix-max scan.

#define T_LEN   2048
#define CHUNK   128                 // 32 lanes * float4
#define NCHUNK  (T_LEN / CHUNK)     // 16
#define WPB     8                   // waves (rows) per 256-thread block
#define LANES   32

// ---- gfx1250 async global->LDS copy (inline CDNA5 asm) ---------------------
// vDst = 32-bit LDS byte address (low 32 bits of the generic shared address),
// vAddr = 64-bit global address. Tracked by ASYNCcnt.
__device__ __forceinline__ void async_ld_b128(unsigned lds_addr, const void* gaddr) {
  asm volatile("global_load_async_to_lds_b128 %0, %1, off"
               :
               : "v"(lds_addr), "v"(gaddr)
               : "memory");
}
__device__ __forceinline__ void s_wait_asynccnt_le2() {
  asm volatile("s_wait_asynccnt 0x2" ::: "memory");
}
__device__ __forceinline__ void s_wait_asynccnt_0() {
  asm volatile("s_wait_asynccnt 0x0" ::: "memory");
}

// Fast tanh: tanh(x) = (e^{2x}-1)/(e^{2x}+1) with HW exp + fast divide.
__device__ __forceinline__ float fast_tanhf(float x) {
  x = fminf(fmaxf(x, -15.0f), 15.0f);
  const float e = __expf(2.0f * x);
  return __fdividef(e - 1.0f, e + 1.0f);
}

__global__ void mdd_init_kernel(unsigned* out_bits) {
  if (threadIdx.x == 0 && blockIdx.x == 0) out_bits[0] = 0u;  // 0.0f
}

__global__ __launch_bounds__(WPB * LANES)
void mdd_scan_kernel(const float* __restrict__ pred,
                     const float* __restrict__ act,
                     unsigned* __restrict__ out_bits,
                     int nrows) {
  // [wave][buf][arr(pred/act)][elem]; each wave owns its slice -> no barriers.
  __shared__ __align__(16) float lds[WPB][2][2][CHUNK];

  const int lane = (int)(threadIdx.x & (LANES - 1));
  const int wave = (int)(threadIdx.x >> 5);
  const int row  = (int)blockIdx.x * WPB + wave;
  if (row >= nrows) return;  // whole wave uniform

  const float* __restrict__ pr = pred + (size_t)row * T_LEN;
  const float* __restrict__ ar = act  + (size_t)row * T_LEN;

  // Low 32 bits of the generic shared-aperture address == LDS byte offset.
  const unsigned lds_base = (unsigned)(uintptr_t)(&lds[wave][0][0][0]);
  const unsigned lane_b   = (unsigned)lane * 16u;

  // Kick off chunk 0 (2 async b128 per chunk: pred + act, 512 B each).
  async_ld_b128(lds_base + 0u * (CHUNK * 4u) + lane_b, pr + lane * 4);
  async_ld_b128(lds_base + 1u * (CHUNK * 4u) + lane_b, ar + lane * 4);

  float carry_prod = 1.0f;        // cumulative product entering this chunk
  float carry_max  = -INFINITY;   // running max entering this chunk
  float dd_max     = 0.0f;

  for (int c = 0; c < NCHUNK; ++c) {
    const int buf = c & 1;
    if (c + 1 < NCHUNK) {
      const int nb = buf ^ 1;
      async_ld_b128(lds_base + (unsigned)((nb * 2 + 0) * CHUNK * 4) + lane_b,
                    pr + (c + 1) * CHUNK + lane * 4);
      async_ld_b128(lds_base + (unsigned)((nb * 2 + 1) * CHUNK * 4) + lane_b,
                    ar + (c + 1) * CHUNK + lane * 4);
      s_wait_asynccnt_le2();      // chunk c done; next-chunk loads in flight
    } else {
      s_wait_asynccnt_0();
    }

    const float4 p4 = *(const float4*)(&lds[wave][buf][0][lane * 4]);
    const float4 a4 = *(const float4*)(&lds[wave][buf][1][lane * 4]);

    // f_i = 1 + tanh(pred_i) * act_i
    const float f0 = fmaf(fast_tanhf(p4.x), a4.x, 1.0f);
    const float f1 = fmaf(fast_tanhf(p4.y), a4.y, 1.0f);
    const float f2 = fmaf(fast_tanhf(p4.z), a4.z, 1.0f);
    const float f3 = fmaf(fast_tanhf(p4.w), a4.w, 1.0f);

    // per-lane inclusive prefix products
    const float l0 = f0;
    const float l1 = l0 * f1;
    const float l2 = l1 * f2;
    const float l3 = l2 * f3;

    // wave32 inclusive product scan of lane totals (5 shuffle steps)
    float incl = l3;
#pragma unroll
    for (int d = 1; d < LANES; d <<= 1) {
      const float o = __shfl_up(incl, (unsigned)d, LANES);
      if (lane >= d) incl *= o;
    }
    float excl = __shfl_up(incl, 1u, LANES);
    if (lane == 0) excl = 1.0f;

    // cumulative returns for this lane's 4 elements
    const float base = carry_prod * excl;
    const float c0 = base * l0;
    const float c1 = base * l1;
    const float c2 = base * l2;
    const float c3 = base * l3;

    // per-lane inclusive prefix max
    const float m0 = c0;
    const float m1 = fmaxf(m0, c1);
    const float m2 = fmaxf(m1, c2);
    const float m3 = fmaxf(m2, c3);

    // wave32 inclusive max scan of lane maxima
    float minc = m3;
#pragma unroll
    for (int d = 1; d < LANES; d <<= 1) {
      const float o = __shfl_up(minc, (unsigned)d, LANES);
      if (lane >= d) minc = fmaxf(minc, o);
    }
    const float mexc  = __shfl_up(minc, 1u, LANES);
    const float rbase = (lane == 0) ? carry_max : fmaxf(carry_max, mexc);

    // running max at each element
    const float r0 = fmaxf(rbase, m0);
    const float r1 = fmaxf(rbase, m1);
    const float r2 = fmaxf(rbase, m2);
    const float r3 = fmaxf(rbase, m3);

    // drawdowns
    float dd = __fdividef(r0 - c0, r0 + 1e-8f);
    dd = fmaxf(dd, __fdividef(r1 - c1, r1 + 1e-8f));
    dd = fmaxf(dd, __fdividef(r2 - c2, r2 + 1e-8f));
    dd = fmaxf(dd, __fdividef(r3 - c3, r3 + 1e-8f));
    dd_max = fmaxf(dd_max, dd);

    // carries: lane 31 holds whole-chunk cumulative product / running max
    carry_prod = __shfl(c3, LANES - 1, LANES);
    carry_max  = __shfl(r3, LANES - 1, LANES);
  }

  // wave max-reduce, then one atomic per row
#pragma unroll
  for (int d = LANES / 2; d > 0; d >>= 1)
    dd_max = fmaxf(dd_max, __shfl_xor(dd_max, (unsigned)d, LANES));

  if (lane == 0) {
    // drawdowns are >= 0: non-negative IEEE floats order as u32 bit patterns
    atomicMax(out_bits, __float_as_uint(dd_max));
  }
}

extern "C" void kernel_launch(void* const* d_in, const int* in_sizes, int n_in,
                              void* d_out, int out_size, void* d_ws, size_t ws_size,
                              hipStream_t stream) {
  (void)n_in; (void)out_size; (void)d_ws; (void)ws_size;
  const float* pred = (const float*)d_in[0];
  const float* act  = (const float*)d_in[1];
  unsigned* out_bits = (unsigned*)d_out;

  const int B = in_sizes[0] / T_LEN;  // 4096 for the reference shapes

  mdd_init_kernel<<<1, 1, 0, stream>>>(out_bits);

  const int blocks = (B + WPB - 1) / WPB;
  mdd_scan_kernel<<<blocks, WPB * LANES, 0, stream>>>(pred, act, out_bits, B);
}